// MultibandResonanceLookup_67310727463592
// MI455X (gfx1250) — compile-verified
//
#include <hip/hip_runtime.h>
#include <hip/hip_bf16.h>
#include <math.h>

// ---------------------------------------------------------------------------
// MultibandResonanceLookup for MI455X (gfx1250, wave32, WGP w/ 320KB LDS)
//
//  K0: transpose+pad items -> WT[608x512] (removes all GEMM bounds checks)
//  K1: WMMA fp32 GEMM  selected[512x608] = relu(selections) @ items
//  K2: per (item,band): synth frames -> OLA -> packed-real FFT(2S) in LDS
//      (LDS twiddle table, zero transcendentals in butterflies)
//      -> masked ortho bins accumulated into per-item spectrum held in d_out
//  K3: per item: TDM async tensor-load of the 128KB half-spectrum into LDS,
//      Hermitian expansion, 32768-pt complex FFT in LDS (irfft(65536) via
//      half-size trick), crop to 32768 samples, fused L2 normalization.
// ---------------------------------------------------------------------------

#define TPB 256
#define PI_F  3.14159265358979323846f
#define TWO_PI_F 6.28318530717958647692f

typedef float v2f  __attribute__((ext_vector_type(2)));
typedef float v8f  __attribute__((ext_vector_type(8)));
typedef unsigned int u32x4 __attribute__((ext_vector_type(4)));
typedef int   i32x4 __attribute__((ext_vector_type(4)));
typedef int   i32x8 __attribute__((ext_vector_type(8)));

#if defined(__has_builtin)
#  if __has_builtin(__builtin_amdgcn_tensor_load_to_lds) && \
      __has_builtin(__builtin_amdgcn_s_wait_tensorcnt)
#    define USE_TDM 1
#  else
#    define USE_TDM 0
#  endif
#else
#  define USE_TDM 0
#endif

// ---------------- K0: transpose + zero-pad items to WT[608][512] -----------
__global__ __launch_bounds__(TPB)
void transpose_pad_items(const float* __restrict__ items,  // 512 x 594
                         float* __restrict__ WT)           // 608 x 512
{
    const int idx = blockIdx.x * TPB + threadIdx.x;        // 608*512
    if (idx >= 608 * 512) return;
    const int col = idx >> 9;         // 0..607
    const int k   = idx & 511;        // 0..511
    WT[idx] = (col < 594) ? items[k * 594 + col] : 0.0f;
}

// ------------------------- K1: WMMA GEMM (fp32) ----------------------------
// selected[512x608] = relu(A[512x512]) * W[512x594->608]. One wave per tile.
// V_WMMA_F32_16X16X4_F32: A 16x4 (lanes 0-15 K={0,1}, 16-31 K={2,3} across
// 2 VGPRs), B 4x16 mirrored, C/D 8 VGPRs (M=r / M=r+8 split at lane 16).
__global__ __launch_bounds__(32)
void wmma_gemm_relu(const float* __restrict__ A,    // selections 512x512
                    const float* __restrict__ WT,   // items^T padded 608x512
                    float* __restrict__ Cout)       // selected 512x608
{
    const int tn   = blockIdx.x;          // 0..37
    const int tm   = blockIdx.y;          // 0..31
    const int lane = threadIdx.x;         // wave32
    const int l    = lane & 15;
    const int hi   = lane >> 4;           // 0: K pair {0,1}, 1: K pair {2,3}
    const int row  = tm * 16 + l;
    const int col  = tn * 16 + l;

    const float* arow = A  + row * 512;
    const float* bcol = WT + col * 512;   // transposed: contiguous in K

    v8f acc = {};
    for (int k = 0; k < 512; k += 4) {
        const int kk = k + hi * 2;
        const float2 av = *(const float2*)(arow + kk);
        const float2 bv = *(const float2*)(bcol + kk);
        v2f a; a.x = fmaxf(av.x, 0.0f); a.y = fmaxf(av.y, 0.0f);
        v2f b; b.x = bv.x;              b.y = bv.y;
        acc = __builtin_amdgcn_wmma_f32_16x16x4_f32(
            false, a, false, b, (short)0, acc, false, false);
    }
    #pragma unroll
    for (int r = 0; r < 8; ++r) {
        const int orow = tm * 16 + hi * 8 + r;     // C/D layout: M=r / M=r+8
        Cout[orow * 608 + col] = acc[r];
    }
}

// ----------------- in-place complex FFT (LDS twiddle table) ----------------
// twd[k] = e^{-2*pi*i*k/L}, k < L/2; stage-s twiddle = twd[pos << (LOG-s)].
__device__ __forceinline__
void fft_lds_tw(float2* buf, const float2* twd, int L, int LOG, int tid)
{
    for (int i = tid; i < L; i += TPB) {
        int r = (int)(__brev((unsigned)i) >> (32 - LOG));
        if (r > i) { float2 t = buf[i]; buf[i] = buf[r]; buf[r] = t; }
    }
    __syncthreads();
    for (int s = 1; s <= LOG; ++s) {
        const int half = 1 << (s - 1);
        const int tsh  = LOG - s;
        for (int b = tid; b < (L >> 1); b += TPB) {
            const int grp = b >> (s - 1);
            const int pos = b & (half - 1);
            const int i0  = (grp << s) + pos;
            const int i1  = i0 + half;
            const float2 w  = twd[pos << tsh];
            const float2 vb = buf[i1];
            const float tr = w.x * vb.x - w.y * vb.y;
            const float ti = w.x * vb.y + w.y * vb.x;
            const float2 va = buf[i0];
            buf[i1] = make_float2(va.x - tr, va.y - ti);
            buf[i0] = make_float2(va.x + tr, va.y + ti);
        }
        __syncthreads();
    }
}

// ------------ in-place complex FFT (sincos twiddles, for L=32768) ----------
__device__ __forceinline__
void fft_lds(float2* buf, int L, int LOG, int tid)
{
    for (int i = tid; i < L; i += TPB) {
        int r = (int)(__brev((unsigned)i) >> (32 - LOG));
        if (r > i) { float2 t = buf[i]; buf[i] = buf[r]; buf[r] = t; }
    }
    __syncthreads();
    for (int s = 1; s <= LOG; ++s) {
        const int half = 1 << (s - 1);
        const float astep = -TWO_PI_F / (float)(1 << s);
        for (int b = tid; b < (L >> 1); b += TPB) {
            const int grp = b >> (s - 1);
            const int pos = b & (half - 1);
            const int i0  = (grp << s) + pos;
            const int i1  = i0 + half;
            float sn, cs; __sincosf(astep * (float)pos, &sn, &cs);
            const float2 vb = buf[i1];
            const float tr = cs * vb.x - sn * vb.y;
            const float ti = cs * vb.y + sn * vb.x;
            const float2 va = buf[i0];
            buf[i1] = make_float2(va.x - tr, va.y - ti);
            buf[i0] = make_float2(va.x + tr, va.y + ti);
        }
        __syncthreads();
    }
}

// ------------------- K2: band synthesis + forward rfft ---------------------
__global__ __launch_bounds__(TPB)
void band_synth(const float* __restrict__ selected,  // 512 x 608
                float* __restrict__ hacc,            // d_out: 512 x 16384 cplx
                float2* __restrict__ topbin)         // bin 16384 per item
{
    const int band = blockIdx.x;          // 0..5
    const int item = blockIdx.y;          // 0..511
    const int S    = 512 << band;         // packed FFT length; real len n = 2S
    const int n    = 2 * S;
    const int F    = S >> 5;              // frames = S/32
    const int LOG  = 9 + band;
    const int tid  = threadIdx.x;

    __shared__ float2 cbuf[16384];        // 128 KB: signal / packed spectrum
    __shared__ float2 twd[8192];          //  64 KB: FFT twiddles e^{-2pi i k/S}
    __shared__ float2 tab[33 * 64];       //  17 KB: cos/sin(2*pi*m*u/64)
    __shared__ float  lmag[33], ph[33], st[33];
    __shared__ float  coefR[8][33], coefI[8][33];
    float* xf = reinterpret_cast<float*>(cbuf);   // 2S reals, aliased

    const float* p = selected + item * 608 + band * 99;
    __builtin_prefetch(p, 0, 1);                  // -> global_prefetch_b8

    // mag = 0.2 + sigmoid*0.8*0.9999, phase = tanh*pi, start = sigmoid
    for (int c = tid; c < 33; c += TPB) {
        const float m = 0.2f + (1.0f / (1.0f + __expf(-p[c]))) * (0.8f * 0.9999f);
        lmag[c] = __logf(m);
        ph[c]   = tanhf(p[33 + c]) * PI_F;
        st[c]   = 1.0f / (1.0f + __expf(-p[66 + c]));
    }
    for (int e = tid; e < 33 * 64; e += TPB) {    // 64-pt DFT basis (+Hann row)
        const int m = e >> 6, u = e & 63;
        float sn, cs; __sincosf(TWO_PI_F * (float)(m * u) * (1.0f / 64.0f), &sn, &cs);
        tab[e] = make_float2(cs, sn);
    }
    for (int k2 = tid; k2 < (S >> 1); k2 += TPB) {  // twiddles, once per block
        float sn, cs; __sincosf(-TWO_PI_F * (float)k2 / (float)S, &sn, &cs);
        twd[k2] = make_float2(cs, sn);
    }
    for (int i = tid; i < 2 * S; i += TPB) xf[i] = 0.0f;
    __syncthreads();

    // ---- frame synthesis + Hann + overlap-add (hop 32) ----
    // s_f(u) = (1/64)[ar0 + 2*sum_{m=1..31}(ar_m*cos - ai_m*sin) + ar32*(-1)^u]
    // ar = amp*cos(phi*f), ai = amp*sin(phi*f), amp = start*mag^f
    const int chunks = F >> 3;            // F in {16..512}, divisible by 8
    for (int ch = 0; ch < chunks; ++ch) {
        for (int e = tid; e < 8 * 33; e += TPB) {
            const int fl = e / 33, m = e - fl * 33;
            const float f = (float)(ch * 8 + fl + 1);
            const float amp = st[m] * __expf(f * lmag[m]);
            float sn, cs; __sincosf(ph[m] * f, &sn, &cs);
            coefR[fl][m] = amp * cs;
            coefI[fl][m] = amp * sn;
        }
        __syncthreads();
        for (int e = tid; e < 8 * 64; e += TPB) {
            const int fl = e >> 6, u = e & 63;
            float acc = coefR[fl][0] + coefR[fl][32] * ((u & 1) ? -1.0f : 1.0f);
            #pragma unroll 8
            for (int m = 1; m < 32; ++m) {
                const float2 t = tab[(m << 6) + u];
                acc += 2.0f * (coefR[fl][m] * t.x - coefI[fl][m] * t.y);
            }
            const float hann = 0.5f * (1.0f - tab[64 + u].x);   // m=1 row
            const float val  = acc * (1.0f / 64.0f) * hann;
            const int f0 = ch * 8 + fl;
            atomicAdd(&xf[f0 * 32 + u], val);   // <=2 writers: deterministic
        }
        __syncthreads();
    }

    // ---- rfft(n=2S, ortho) via packed complex FFT of length S ----
    fft_lds_tw(cbuf, twd, S, LOG, tid);

    // X[j] = 0.5[(C[j]+conj(C[S-j])) - i*e^{-i*pi*j/S}*(C[j]-conj(C[S-j]))]
    const float scale = 1.0f / sqrtf((float)n);   // ortho
    const int jlo = (band == 0) ? 0 : (S >> 1);   // mask: [0,S] or [S/2,S]
    for (int j = jlo + tid; j <= S; j += TPB) {
        float2 Cj = (j == S) ? cbuf[0] : cbuf[j];
        float2 Ck = (j == 0) ? cbuf[0] : cbuf[S - j];
        Ck.y = -Ck.y;                               // conj
        const float Sr = Cj.x + Ck.x, Si = Cj.y + Ck.y;
        const float Dr = Cj.x - Ck.x, Di = Cj.y - Ck.y;
        float sn, cs; __sincosf(-PI_F * (float)j / (float)S, &sn, &cs);
        const float iwDr = -(cs * Di + sn * Dr);    // i*w*D, w = e^{-i*pi*j/S}
        const float iwDi =  (cs * Dr - sn * Di);
        const float Xr = 0.5f * (Sr - iwDr) * scale;
        const float Xi = 0.5f * (Si - iwDi) * scale;
        if (j == 16384) {
            topbin[item] = make_float2(Xr, Xi);     // single writer (band 5)
        } else {
            atomicAdd(&hacc[item * 32768 + 2 * j],     Xr);  // <=2 adds onto 0:
            atomicAdd(&hacc[item * 32768 + 2 * j + 1], Xi);  // deterministic
        }
    }
}

// ------------- K3: irfft(65536, ortho) + crop + normalize ------------------
// C[k] = 0.5*H[k]*((1-sin)+i*cos), C[M-k] = 0.5*conj(H[k])*((1+sin)+i*cos),
// C[0] = 0.5*H0*(1+i), C[M/2] = conj(H[M/2]); M=32768.
// z = (sqrt(N)/M)*conj(FFT(conj(C))), samples interleaved z[2t], z[2t+1].
__global__ __launch_bounds__(TPB)
void spectrum_finish(float* __restrict__ io,            // d_out (in: H, out: z)
                     const float2* __restrict__ topbin)
{
    const int item = blockIdx.x;
    const int tid  = threadIdx.x;
    __shared__ float2 BUF[32768];          // 256 KB (gfx1250 WGP LDS)
    __shared__ float  red[TPB];

    const float2* hin = reinterpret_cast<const float2*>(io) + (size_t)item * 16384;

#if USE_TDM
    // Tensor Data Mover: DMA the 16384x8B half-spectrum straight into LDS.
    // D# per CDNA5 ISA ch.8: group0 = {count=1 | type=2, lds_addr, gaddr};
    // group1 = {data_size=8B, tensor_dim0=tile_dim0=16384, dims1=1, strides}.
    // This toolchain exposes the 6-arg builtin (extra int32x8 group, zeroed).
    if (tid < 32) {                        // one wave issues the descriptor
        const unsigned long long ga = (unsigned long long)(uintptr_t)hin;
        const unsigned int ldsa = (unsigned int)(uintptr_t)(void*)BUF; // low 32b
        u32x4 g0 = {0u, 0u, 0u, 0u};
        g0[0] = 1u;                                   // count=1 (user mode)
        g0[1] = ldsa;                                 // lds_addr[31:0]
        g0[2] = (unsigned int)(ga & 0xFFFFFFFFull);   // global_addr[31:0]
        g0[3] = (unsigned int)((ga >> 32) & 0x1FFFFFFull) | (2u << 30); // +type=2
        i32x8 g1 = {0, 0, 0, 0, 0, 0, 0, 0};
        g1[0] = (int)(3u << 16);                      // data_size=3 (8 bytes)
        g1[1] = (int)(16384u << 16);                  // tensor_dim0[15:0]
        g1[2] = (int)(1u << 16);                      // dim0 hi=0 | tensor_dim1=1
        g1[3] = (int)(16384u << 16);                  // dim1 hi=0 | tile_dim0
        g1[4] = 1;                                    // tile_dim1=1, tile_dim2=0
        g1[5] = 16384;                                // tensor_dim0_stride lo32
        g1[6] = (int)(16384u << 16);                  // stride0 hi | stride1 lo
        i32x4 g2 = {0, 0, 0, 0}, g3 = {0, 0, 0, 0};
        i32x8 g4 = {0, 0, 0, 0, 0, 0, 0, 0};          // extension group (unused)
        __builtin_amdgcn_tensor_load_to_lds(g0, g1, g2, g3, g4, 0);
        __builtin_amdgcn_s_wait_tensorcnt(0);
    }
    for (int j = 16384 + tid; j < 32768; j += TPB)  BUF[j] = make_float2(0.0f, 0.0f);
#else
    __builtin_prefetch(hin, 0, 1);
    for (int j = tid; j < 16384; j += TPB)          BUF[j] = hin[j];
    for (int j = 16384 + tid; j < 32768; j += TPB)  BUF[j] = make_float2(0.0f, 0.0f);
#endif
    __syncthreads();

    for (int j = tid; j < 16384; j += TPB) {
        if (j == 0) {
            const float h0 = BUF[0].x;               // irfft ignores imag(H0)
            BUF[0] = make_float2(0.5f * h0, -0.5f * h0);        // conj(C[0])
        } else {
            const float2 h = BUF[j];
            float sn, cs; __sincosf(TWO_PI_F * (float)j / 65536.0f, &sn, &cs);
            const float ar = 1.0f - sn, ai = cs;     // C[j]   = 0.5*h*(ar+i*ai)
            BUF[j] = make_float2(0.5f * (h.x * ar - h.y * ai),
                                -0.5f * (h.x * ai + h.y * ar));      // conj
            const float br = 1.0f + sn, bi = cs;     // C[M-j] = 0.5*conj(h)*(br+i*bi)
            const float hr = h.x, hi2 = -h.y;
            BUF[32768 - j] = make_float2(0.5f * (hr * br - hi2 * bi),
                                        -0.5f * (hr * bi + hi2 * br)); // conj
        }
    }
    if (tid == 0) BUF[16384] = topbin[item];          // conj(conj(h)) = h
    __syncthreads();

    fft_lds(BUF, 32768, 15, tid);

    // L2 norm over the 32768 output samples (t < 16384 complex pairs)
    const float k = 256.0f / 32768.0f;                // sqrt(65536)/M
    float part = 0.0f;
    for (int t = tid; t < 16384; t += TPB) {
        const float zr =  BUF[t].x * k;
        const float zi = -BUF[t].y * k;
        part += zr * zr + zi * zi;
    }
    red[tid] = part;
    __syncthreads();
    for (int off = TPB >> 1; off > 0; off >>= 1) {
        if (tid < off) red[tid] += red[tid + off];
        __syncthreads();
    }
    const float factor = 1.0f / (sqrtf(red[0]) + 1e-8f);

    float* zo = io + (size_t)item * 32768;
    for (int t = tid; t < 16384; t += TPB) {
        const float zr =  BUF[t].x * k;
        const float zi = -BUF[t].y * k;
        zo[2 * t]     = zr * factor;
        zo[2 * t + 1] = zi * factor;
    }
}

// ------------------------------- launch ------------------------------------
extern "C" void kernel_launch(void* const* d_in, const int* in_sizes, int n_in,
                              void* d_out, int out_size, void* d_ws, size_t ws_size,
                              hipStream_t stream)
{
    const float* selections = (const float*)d_in[0];   // 8*16*4*512 fp32
    const float* items      = (const float*)d_in[1];   // 512*594 fp32
    float*  out      = (float*)d_out;                  // 512*32768 fp32

    float*  selected = (float*)d_ws;                                    // 512*608
    float*  WT       = selected + (size_t)512 * 608;                    // 608*512
    float2* topbin   = (float2*)(WT + (size_t)608 * 512);               // 512

    // d_out doubles as the per-item spectrum accumulator; zero it (+topbin).
    (void)hipMemsetAsync(d_out, 0, (size_t)512 * 32768 * sizeof(float), stream);
    (void)hipMemsetAsync(topbin, 0, 512 * sizeof(float2), stream);

    transpose_pad_items<<<(608 * 512 + TPB - 1) / TPB, TPB, 0, stream>>>(items, WT);
    wmma_gemm_relu     <<<dim3(38, 32), 32, 0, stream>>>(selections, WT, selected);
    band_synth         <<<dim3(6, 512), TPB, 0, stream>>>(selected, out, topbin);
    spectrum_finish    <<<512, TPB, 0, stream>>>(out, topbin);
}